// g_data_net_87686052315570
// MI455X (gfx1250) — compile-verified
//
#include <hip/hip_runtime.h>
#include <stdint.h>

// Problem constants (from reference): D=W=32, 22 classes, row = 22*32+32+32 = 768 f32.
#define NCLS    22
#define DCOLS   32
#define WCOLS   32
#define ROW_OUT 768
#define OH_F4   176   // one-hot region = 704 floats = 176 float4 chunks
#define ROW_F4  192   // 768 / 4

typedef float v4f __attribute__((ext_vector_type(4)));

// ---------------------------------------------------------------------------
// Kernel A: re-initialize the min/max accumulators (must run every call).
// All gathered dist values are >= 0.0f, so IEEE bit patterns order like uints.
// ws[0] = running min bits (init +inf), ws[1] = running max bits (init 0).
// ---------------------------------------------------------------------------
__global__ void init_ws_kernel(unsigned* ws) {
  ws[0] = 0x7F800000u;  // +inf
  ws[1] = 0u;           // 0.0f
}

// ---------------------------------------------------------------------------
// Kernel B: global min/max over dist_t[h,w] = dist_new[index_h[h], index_t[h,w]].
// Grid-stride; consecutive lanes share one row h (index_h load is broadcast).
// Block tree-reduction in LDS, then one uint atomicMin/atomicMax per block.
// Deterministic: min/max are order-independent.
// ---------------------------------------------------------------------------
__global__ void __launch_bounds__(256)
minmax_kernel(const float* __restrict__ dist,
              const int* __restrict__ index_t,
              const int* __restrict__ index_h,
              unsigned* __restrict__ ws,
              int total /* = H * 32 */) {
  unsigned mn = 0x7F800000u, mx = 0u;
  const int stride = gridDim.x * blockDim.x;
  for (int e = blockIdx.x * blockDim.x + threadIdx.x; e < total; e += stride) {
    const int h = e >> 5;                 // row (32 gathers per row)
    const int j = index_t[e];             // 0..32, 32 == zero sentinel
    const float v = (j < DCOLS) ? dist[(size_t)index_h[h] * DCOLS + j] : 0.0f;
    const unsigned b = __float_as_uint(v);
    mn = (b < mn) ? b : mn;
    mx = (b > mx) ? b : mx;
  }

  __shared__ unsigned smin[256];
  __shared__ unsigned smax[256];
  const int t = threadIdx.x;
  smin[t] = mn;
  smax[t] = mx;
  __syncthreads();
#pragma unroll
  for (int s = 128; s > 0; s >>= 1) {
    if (t < s) {
      const unsigned a0 = smin[t + s];
      const unsigned a1 = smax[t + s];
      if (a0 < smin[t]) smin[t] = a0;
      if (a1 > smax[t]) smax[t] = a1;
    }
    __syncthreads();
  }
  if (t == 0) {
    atomicMin(&ws[0], smin[0]);
    atomicMax(&ws[1], smax[0]);
  }
}

// ---------------------------------------------------------------------------
// Kernel C: emit the [H, 768] output. 256-thread block = 4 row-groups of 64
// lanes. Per row:
//   1) stage idx_t row + index_t row (128 B each) into LDS with the CDNA5
//      async copy engine (global_load_async_to_lds_b128, ASYNCcnt),
//   2) 32 lanes gather + normalize (exact IEEE division, matches reference
//      rounding) dist/angle into LDS,
//   3) 64 lanes store 3 float4 chunks each (192 chunks = 768 floats) with
//      NON-TEMPORAL B128 stores: the 307 MB output is write-once, keep the
//      reused inputs resident in the 192 MB L2 instead.
//      One-hot chunks computed in registers: a 4-wide chunk overlaps at most
//      two 22-wide one-hot segments.
// ---------------------------------------------------------------------------
__global__ void __launch_bounds__(256)
emit_kernel(const float* __restrict__ dist,
            const float* __restrict__ angle,
            const int* __restrict__ idx_t,
            const int* __restrict__ index_t,
            const int* __restrict__ index_h,
            const unsigned* __restrict__ ws,
            float* __restrict__ out,
            int H) {
  __shared__ int   s_idx[4][WCOLS];
  __shared__ int   s_jdx[4][WCOLS];
  __shared__ float s_d[4][WCOLS];
  __shared__ float s_a[4][WCOLS];

  const int t = threadIdx.x;
  const int g = t >> 6;        // row-group within block (0..3)
  const int l = t & 63;        // lane within row-group
  const int h = blockIdx.x * 4 + g;
  const bool valid = (h < H);

  // -- Stage 1: async global->LDS copy of this row's idx/index data ---------
  // Lanes 0..7: 8 x b128 = idx_t row; lanes 8..15: index_t row.
  if (valid && l < 16) {
    unsigned lds_off;
    uint64_t src;
    if (l < 8) {
      lds_off = (unsigned)(uintptr_t)&s_idx[g][l * 4];
      src = (uint64_t)(uintptr_t)(idx_t + (size_t)h * WCOLS + l * 4);
    } else {
      lds_off = (unsigned)(uintptr_t)&s_jdx[g][(l - 8) * 4];
      src = (uint64_t)(uintptr_t)(index_t + (size_t)h * WCOLS + (l - 8) * 4);
    }
    asm volatile("global_load_async_to_lds_b128 %0, %1, off"
                 :: "v"(lds_off), "v"(src)
                 : "memory");
  }
  asm volatile("s_wait_asynccnt 0" ::: "memory");
  __syncthreads();

  // -- Stage 2: gather + normalize into LDS ---------------------------------
  const float dmin  = __uint_as_float(ws[0]);
  const float dmax  = __uint_as_float(ws[1]);
  const float range = dmax - dmin;

  if (valid && l < WCOLS) {
    const int rh = index_h[h];
    const int j = s_jdx[g][l];
    float dv = 0.0f, av = 0.0f;
    if (j < DCOLS) {
      dv = dist[(size_t)rh * DCOLS + j];
      av = angle[(size_t)rh * DCOLS + j];
    }
    s_d[g][l] = (dv - dmin) / range;   // exact IEEE divide, matches reference
    s_a[g][l] = av;
  }
  __syncthreads();

  // -- Stage 3: coalesced non-temporal float4 stores of the 768-float row ---
  if (valid) {
    v4f* __restrict__ orow = (v4f*)(out + (size_t)h * ROW_OUT);
#pragma unroll
    for (int i = 0; i < 3; ++i) {
      const int c = l + 64 * i;           // chunk 0..191
      v4f v;
      if (c < OH_F4) {
        // one-hot: column span [base, base+3]; overlaps segments w0, w1.
        const int base = c * 4;
        const int w0 = base / NCLS;
        const int w1 = (base + 3) / NCLS;         // w1 == w0 or w0+1, <= 31
        const int p0 = w0 * NCLS + s_idx[g][w0];  // hot column in segment w0
        const int p1 = w1 * NCLS + s_idx[g][w1];  // hot column in segment w1
        v.x = (p0 == base     || p1 == base    ) ? 1.0f : 0.0f;
        v.y = (p0 == base + 1 || p1 == base + 1) ? 1.0f : 0.0f;
        v.z = (p0 == base + 2 || p1 == base + 2) ? 1.0f : 0.0f;
        v.w = (p0 == base + 3 || p1 == base + 3) ? 1.0f : 0.0f;
      } else if (c < OH_F4 + 8) {
        const int e = (c - OH_F4) * 4;
        v.x = s_d[g][e];     v.y = s_d[g][e + 1];
        v.z = s_d[g][e + 2]; v.w = s_d[g][e + 3];
      } else {
        const int e = (c - (OH_F4 + 8)) * 4;
        v.x = s_a[g][e];     v.y = s_a[g][e + 1];
        v.z = s_a[g][e + 2]; v.w = s_a[g][e + 3];
      }
      __builtin_nontemporal_store(v, &orow[c]);   // global_store_b128 ... NT
    }
  }
}

// ---------------------------------------------------------------------------
// Host-side launcher. Inputs (setup_inputs order):
//   d_in[0] = dist    [N,32] f32
//   d_in[1] = angle   [N,32] f32
//   d_in[2] = idx_t   [H,32] i32
//   d_in[3] = index_t [H,32] i32
//   d_in[4] = index_h [H]    i32
// Output: [H, 768] f32.
// ---------------------------------------------------------------------------
extern "C" void kernel_launch(void* const* d_in, const int* in_sizes, int n_in,
                              void* d_out, int out_size, void* d_ws, size_t ws_size,
                              hipStream_t stream) {
  const float* dist    = (const float*)d_in[0];
  const float* angle   = (const float*)d_in[1];
  const int*   idx_t   = (const int*)d_in[2];
  const int*   index_t = (const int*)d_in[3];
  const int*   index_h = (const int*)d_in[4];
  float*       out     = (float*)d_out;
  unsigned*    ws      = (unsigned*)d_ws;

  const int H = in_sizes[4];           // 100000
  const int total = H * WCOLS;         // 3.2M gathered elements

  init_ws_kernel<<<1, 1, 0, stream>>>(ws);

  int rblocks = (total + 255) / 256;
  if (rblocks > 1024) rblocks = 1024;  // grid-stride; ~12 elems/thread
  minmax_kernel<<<rblocks, 256, 0, stream>>>(dist, index_t, index_h, ws, total);

  const int eblocks = (H + 3) / 4;     // 4 rows per 256-thread block
  emit_kernel<<<eblocks, 256, 0, stream>>>(dist, angle, idx_t, index_t,
                                           index_h, ws, out, H);
}